// HGNN_conv_87471303950814
// MI455X (gfx1250) — compile-verified
//
#include <hip/hip_runtime.h>
#include <stdint.h>

#define NN 16384   // nodes
#define NE 8192    // hyperedges
#define FT 64      // in/out features

typedef __attribute__((ext_vector_type(2))) float v2f;
typedef __attribute__((ext_vector_type(4))) float v4f;
typedef __attribute__((ext_vector_type(8))) float v8f;
typedef __attribute__((ext_vector_type(4))) int   v4i;

// ---------------------------------------------------------------------------
// Async global->LDS staging (CDNA5 GLOBAL_LOAD_ASYNC_TO_LDS_B128, ASYNCcnt).
// Builtin signature (probe-confirmed by frontend): (int4 __global* src,
// int4 LDS* dst, imm offset, imm cpol). Falls back to sync copy if absent.
// ---------------------------------------------------------------------------
#if defined(__HIP_DEVICE_COMPILE__) && __has_builtin(__builtin_amdgcn_global_load_async_to_lds_b128)
#define USE_ASYNC 1
#else
#define USE_ASYNC 0
#endif

typedef __attribute__((address_space(1))) v4i gas_v4i;
typedef __attribute__((address_space(3))) v4i las_v4i;

__device__ __forceinline__ void async_ld_b128(const float* g, float* l) {
#if USE_ASYNC
  // generic LDS address: low 32 bits are the LDS byte offset (aperture rule)
  __builtin_amdgcn_global_load_async_to_lds_b128(
      (gas_v4i*)(uintptr_t)g,
      (las_v4i*)(unsigned)(uintptr_t)l,
      0, 0);
#else
  *(v4f*)l = *(const v4f*)g;
#endif
}

__device__ __forceinline__ void async_wait0() {
#if USE_ASYNC
#if __has_builtin(__builtin_amdgcn_s_wait_asynccnt)
  __builtin_amdgcn_s_wait_asynccnt(0);
#else
  asm volatile("s_wait_asynccnt 0" ::: "memory");
#endif
#endif
}

// ---------------------------------------------------------------------------
// Kernel 1: one pass over H producing exact row sums (-> dv = rsqrt) and
// per-strip column partial sums (deterministic, no atomics).
// ---------------------------------------------------------------------------
__global__ __launch_bounds__(512) void k_sums(const float* __restrict__ H,
                                              float* __restrict__ colpart,
                                              float* __restrict__ dv) {
  __shared__ float tile[64][129];
  const int strip = blockIdx.x;             // 0..255 (64-row strips)
  const int r0 = strip * 64;
  const int t = threadIdx.x;                // 0..511
  float rowacc = 0.f;
  for (int ct = 0; ct < 64; ++ct) {         // 64 column tiles of 128
    const int c0 = ct * 128;
#pragma unroll
    for (int i = 0; i < 4; ++i) {
      int lin = (t + 512 * i) * 4;
      int r = lin >> 7, c = lin & 127;
      v4f v = *(const v4f*)(H + (size_t)(r0 + r) * NE + c0 + c);
      tile[r][c]     = v.x; tile[r][c + 1] = v.y;
      tile[r][c + 2] = v.z; tile[r][c + 3] = v.w;
    }
    __syncthreads();
    if (t < 64) {
      float s = 0.f;
#pragma unroll
      for (int c = 0; c < 128; ++c) s += tile[t][c];
      rowacc += s;
    } else if (t < 192) {
      int c = t - 64;
      float s = 0.f;
#pragma unroll
      for (int r = 0; r < 64; ++r) s += tile[r][c];
      colpart[(size_t)strip * NE + c0 + c] = s;
    }
    __syncthreads();
  }
  if (t < 64) dv[r0 + t] = rsqrtf(rowacc);
}

// ---------------------------------------------------------------------------
// Kernel 2: dew[e] = W[e] / colsum[e]   (folds diag(W) * DE^-1)
// ---------------------------------------------------------------------------
__global__ void k_dew(const float* __restrict__ colpart,
                      const float* __restrict__ W,
                      float* __restrict__ dew) {
  int e = blockIdx.x * 256 + threadIdx.x;
  float s = 0.f;
  for (int k = 0; k < 256; ++k) s += colpart[(size_t)k * NE + e];
  dew[e] = W[e] / s;
}

// ---------------------------------------------------------------------------
// Kernel 3: xs = dv (.) (x @ weight), WMMA f32 16x16x4, weight in LDS.
// ---------------------------------------------------------------------------
__global__ __launch_bounds__(256) void k_xw(const float* __restrict__ x,
                                            const float* __restrict__ w,
                                            const float* __restrict__ dv,
                                            float* __restrict__ xs) {
  __shared__ float wl[64 * 64];
  const int t = threadIdx.x;
#pragma unroll
  for (int i = 0; i < 16; ++i) wl[t + 256 * i] = w[t + 256 * i];
  __syncthreads();
  const int wave = t >> 5, lane = t & 31;
  const int row0 = blockIdx.x * 128 + wave * 16;
  const int mrow = lane & 15;
  const int kd = (lane < 16) ? 0 : 2;
  v8f acc[4] = {};
#pragma unroll 4
  for (int k0 = 0; k0 < 64; k0 += 4) {
    const int kr = k0 + kd;
    v2f a = *(const v2f*)(x + (size_t)(row0 + mrow) * FT + kr);
#pragma unroll
    for (int c = 0; c < 4; ++c) {
      v2f b;
      b.x = wl[kr * 64 + c * 16 + mrow];
      b.y = wl[(kr + 1) * 64 + c * 16 + mrow];
      acc[c] = __builtin_amdgcn_wmma_f32_16x16x4_f32(
          false, a, false, b, (short)0, acc[c], false, false);
    }
  }
  const int hi = (lane < 16) ? 0 : 8;
#pragma unroll
  for (int c = 0; c < 4; ++c)
#pragma unroll
    for (int r = 0; r < 8; ++r) {
      int row = row0 + r + hi;
      xs[(size_t)row * FT + c * 16 + mrow] = acc[c][r] * dv[row];
    }
}

// ---------------------------------------------------------------------------
// Kernel 4: y_partial = H^T @ xs   (M=e, K=n). H streamed exactly once via
// async-to-LDS double buffering. Dynamic LDS: H tiles 2x(64*128) + B 2x(64*64).
// ---------------------------------------------------------------------------
#define EDGE_HT (64 * 128)
#define TILE_B  (64 * 64)

__global__ __launch_bounds__(256) void k_edge(const float* __restrict__ H,
                                              const float* __restrict__ xs,
                                              float* __restrict__ part) {
  extern __shared__ float smem[];
  float* const Hbase = smem;                  // 2 * EDGE_HT
  float* const Bbase = smem + 2 * EDGE_HT;    // 2 * TILE_B
  const int t = threadIdx.x;
  const int wave = t >> 5, lane = t & 31;
  const int eb = blockIdx.x * 128;
  const int kbase = blockIdx.y * 2048;
  const int mrow = lane & 15;
  const int wcol = wave * 16 + mrow;
  const int kd = (lane < 16) ? 0 : 2;
  v8f acc[4] = {};

  auto stage = [&](int kc, int bu) {
    float* hd = Hbase + bu * EDGE_HT;
    float* bd = Bbase + bu * TILE_B;
#pragma unroll
    for (int i = 0; i < 8; ++i) {     // H tile 64 x 128 (b128 lines)
      int lin = (t + 256 * i) * 4;
      int r = lin >> 7, c = lin & 127;
      async_ld_b128(H + (size_t)(kbase + kc + r) * NE + eb + c, hd + r * 128 + c);
    }
#pragma unroll
    for (int i = 0; i < 4; ++i) {     // xs tile 64 x 64
      int lin = (t + 256 * i) * 4;
      int r = lin >> 6, c = lin & 63;
      async_ld_b128(xs + (size_t)(kbase + kc + r) * FT + c, bd + r * 64 + c);
    }
  };

  stage(0, 0);
  async_wait0();
  __syncthreads();
  int cur = 0;
  for (int kc = 0; kc < 2048; kc += 64) {
    if (kc + 64 < 2048) stage(kc + 64, cur ^ 1);
    const float* hb = Hbase + cur * EDGE_HT;
    const float* bb = Bbase + cur * TILE_B;
#pragma unroll 4
    for (int k0 = 0; k0 < 64; k0 += 4) {
      const int kr = k0 + kd;
      v2f a;
      a.x = hb[kr * 128 + wcol];
      a.y = hb[(kr + 1) * 128 + wcol];
#pragma unroll
      for (int c = 0; c < 4; ++c) {
        v2f b;
        b.x = bb[kr * 64 + c * 16 + mrow];
        b.y = bb[(kr + 1) * 64 + c * 16 + mrow];
        acc[c] = __builtin_amdgcn_wmma_f32_16x16x4_f32(
            false, a, false, b, (short)0, acc[c], false, false);
      }
    }
    async_wait0();      // my async fills for next buffer are done
    __syncthreads();    // everyone's fills done; everyone done reading cur
    cur ^= 1;
  }
  const int hi = (lane < 16) ? 0 : 8;
  const int e0 = eb + wave * 16;
  float* outp = part + (size_t)blockIdx.y * ((size_t)NE * FT);
#pragma unroll
  for (int c = 0; c < 4; ++c)
#pragma unroll
    for (int r = 0; r < 8; ++r)
      outp[(size_t)(e0 + r + hi) * FT + c * 16 + mrow] = acc[c][r];
}

// ---------------------------------------------------------------------------
// Kernel 5: y2 = dew (.) sum_k y_partial (deterministic fixed-order reduce)
// ---------------------------------------------------------------------------
__global__ void k_y2(const float* __restrict__ part,
                     const float* __restrict__ dew,
                     float* __restrict__ y2) {
  int idx = blockIdx.x * 256 + threadIdx.x; // < NE*FT
  int e = idx >> 6;
  float s = 0.f;
#pragma unroll
  for (int k = 0; k < 8; ++k) s += part[(size_t)k * ((size_t)NE * FT) + idx];
  y2[idx] = dew[e] * s;
}

// ---------------------------------------------------------------------------
// Kernel 6: out_partial = H @ y2 (M=n, K=e). H tile async-staged to LDS with
// pad-68 rows -> conflict-free transposed A reads, 16B-aligned b128 fills.
// Dynamic LDS: H tiles 2x(128*68) + B tiles 2x(64*64).
// ---------------------------------------------------------------------------
#define NODE_HT (128 * 68)

__global__ __launch_bounds__(256) void k_node(const float* __restrict__ H,
                                              const float* __restrict__ y2,
                                              float* __restrict__ part) {
  extern __shared__ float smem[];
  float* const Hbase = smem;                  // 2 * NODE_HT, row stride 68
  float* const Bbase = smem + 2 * NODE_HT;    // 2 * TILE_B
  const int t = threadIdx.x;
  const int wave = t >> 5, lane = t & 31;
  const int nb = blockIdx.x * 128;
  const int kbase = blockIdx.y * 2048;
  const int mrow = lane & 15;
  const int wrow = wave * 16 + mrow;
  const int kd = (lane < 16) ? 0 : 2;
  v8f acc[4] = {};

  auto stage = [&](int kc, int bu) {
    float* hd = Hbase + bu * NODE_HT;
    float* bd = Bbase + bu * TILE_B;
#pragma unroll
    for (int i = 0; i < 8; ++i) {     // H tile 128 x 64 (b128 lines)
      int lin = (t + 256 * i) * 4;
      int r = lin >> 6, c = lin & 63;
      async_ld_b128(H + (size_t)(nb + r) * NE + kbase + kc + c, hd + r * 68 + c);
    }
#pragma unroll
    for (int i = 0; i < 4; ++i) {     // y2 tile 64 x 64
      int lin = (t + 256 * i) * 4;
      int r = lin >> 6, c = lin & 63;
      async_ld_b128(y2 + (size_t)(kbase + kc + r) * FT + c, bd + r * 64 + c);
    }
  };

  stage(0, 0);
  async_wait0();
  __syncthreads();
  int cur = 0;
  for (int kc = 0; kc < 2048; kc += 64) {
    if (kc + 64 < 2048) stage(kc + 64, cur ^ 1);
    const float* hb = Hbase + cur * NODE_HT;
    const float* bb = Bbase + cur * TILE_B;
#pragma unroll 4
    for (int k0 = 0; k0 < 64; k0 += 4) {
      const int kr = k0 + kd;
      v2f a = *(const v2f*)(hb + wrow * 68 + kr);   // 8B aligned, banks 4r+kr
#pragma unroll
      for (int c = 0; c < 4; ++c) {
        v2f b;
        b.x = bb[kr * 64 + c * 16 + mrow];
        b.y = bb[(kr + 1) * 64 + c * 16 + mrow];
        acc[c] = __builtin_amdgcn_wmma_f32_16x16x4_f32(
            false, a, false, b, (short)0, acc[c], false, false);
      }
    }
    async_wait0();
    __syncthreads();
    cur ^= 1;
  }
  const int hi = (lane < 16) ? 0 : 8;
  const int n0 = nb + wave * 16;
  float* outp = part + (size_t)blockIdx.y * ((size_t)NN * FT);
#pragma unroll
  for (int c = 0; c < 4; ++c)
#pragma unroll
    for (int r = 0; r < 8; ++r)
      outp[(size_t)(n0 + r + hi) * FT + c * 16 + mrow] = acc[c][r];
}

// ---------------------------------------------------------------------------
// Kernel 7: out = dv (.) sum_k out_partial + bias
// ---------------------------------------------------------------------------
__global__ void k_out(const float* __restrict__ part,
                      const float* __restrict__ dv,
                      const float* __restrict__ bias,
                      float* __restrict__ out) {
  int idx = blockIdx.x * 256 + threadIdx.x; // < NN*FT
  int n = idx >> 6, f = idx & 63;
  float s = 0.f;
#pragma unroll
  for (int k = 0; k < 4; ++k) s += part[(size_t)k * ((size_t)NN * FT) + idx];
  out[idx] = dv[n] * s + bias[f];
}

// ---------------------------------------------------------------------------
extern "C" void kernel_launch(void* const* d_in, const int* in_sizes, int n_in,
                              void* d_out, int out_size, void* d_ws, size_t ws_size,
                              hipStream_t stream) {
  const float* x      = (const float*)d_in[0];   // [NN, FT]
  const float* H      = (const float*)d_in[1];   // [NN, NE]
  const float* W      = (const float*)d_in[2];   // [NE]
  const float* weight = (const float*)d_in[3];   // [FT, FT]
  const float* bias   = (const float*)d_in[4];   // [FT]
  float* out = (float*)d_out;

  // workspace layout (floats), ~30.1 MB total
  float* ws      = (float*)d_ws;
  float* colpart = ws;                                  // 256 * NE
  float* xs      = colpart + (size_t)256 * NE;          // NN * FT
  float* part    = xs + (size_t)NN * FT;                // 8*NE*FT == 4*NN*FT
  float* y2      = part + (size_t)8 * NE * FT;          // NE * FT
  float* dv      = y2 + (size_t)NE * FT;                // NN
  float* dew     = dv + NN;                             // NE

  const size_t lds_edge = (size_t)(2 * EDGE_HT + 2 * TILE_B) * sizeof(float); // 96 KB
  const size_t lds_node = (size_t)(2 * NODE_HT + 2 * TILE_B) * sizeof(float); // 100 KB

  k_sums<<<256, 512, 0, stream>>>(H, colpart, dv);
  k_dew <<<NE / 256, 256, 0, stream>>>(colpart, W, dew);
  k_xw  <<<NN / 128, 256, 0, stream>>>(x, weight, dv, xs);
  k_edge<<<dim3(NE / 128, 8), 256, lds_edge, stream>>>(H, xs, part);
  k_y2  <<<(NE * FT) / 256, 256, 0, stream>>>(part, dew, y2);
  k_node<<<dim3(NN / 128, 4), 256, lds_node, stream>>>(H, y2, part);
  k_out <<<(NN * FT) / 256, 256, 0, stream>>>(part, dv, bias, out);
}